// PatCRF_54056458387646
// MI455X (gfx1250) — compile-verified
//
#include <hip/hip_runtime.h>
#include <hip/hip_bf16.h>

#define BATCH 256
#define SEQ   1024
#define NTAGS 64
#define NLAB  32
#define NPAT  4
#define NEG_INF -10000.0f
#define LN2F   0.69314718056f
#define INVLN2 1.44269504089f

typedef __attribute__((ext_vector_type(2))) float v2f;
typedef __attribute__((ext_vector_type(8))) float v8f;

// Raw hardware log2/exp2 (v_log_f32 / v_exp_f32).
__device__ __forceinline__ float hw_log2(float x) { return __builtin_amdgcn_logf(x); }
__device__ __forceinline__ float hw_exp2(float x) { return __builtin_amdgcn_exp2f(x); }

// ---------------------------------------------------------------------------
// Phase 0: T[i][j] = label_transitions[t2l[i], t2l[j]] + constraints[i][j],
// E = exp(T) (exp(-1e4) -> exactly 0 in fp32, which is what we want).
// ---------------------------------------------------------------------------
__global__ void crf_build_trans(const float* __restrict__ lt,
                                const int* __restrict__ t2l,
                                const float* __restrict__ tc,
                                float* __restrict__ E,
                                float* __restrict__ T) {
    int idx = blockIdx.x * blockDim.x + threadIdx.x;   // 0..4095
    int i = idx >> 6, j = idx & 63;
    float t = lt[t2l[i] * NLAB + t2l[j]] + tc[idx];
    T[idx] = t;
    E[idx] = expf(t);
}

// ---------------------------------------------------------------------------
// Phase 1: score(b) = sum_s x[b,s,y[b,s]] + sum_s T[y[s-1],y[s]]
// ---------------------------------------------------------------------------
__global__ void crf_score(const float* __restrict__ labE,
                          const float* __restrict__ patE,
                          const int* __restrict__ y,
                          const int* __restrict__ t2l,
                          const float* __restrict__ tp,
                          const float* __restrict__ T,
                          float* __restrict__ score) {
    int b = blockIdx.x;
    int tid = threadIdx.x;
    float acc = 0.f;
    for (int s = tid; s < SEQ; s += 256) {
        int yt = y[b * SEQ + s];
        const float* le = labE + (((size_t)b * SEQ + s) * NLAB);
        const float* pe = patE + (((size_t)b * SEQ + s) * NPAT);
        float e = le[t2l[yt]];
        e += pe[0] * tp[yt * 4 + 0] + pe[1] * tp[yt * 4 + 1]
           + pe[2] * tp[yt * 4 + 2] + pe[3] * tp[yt * 4 + 3];
        acc += e;
        if (s > 0) acc += T[y[b * SEQ + s - 1] * NTAGS + yt];
    }
    __shared__ float red[256];
    red[tid] = acc;
    __syncthreads();
    for (int off = 128; off > 0; off >>= 1) {
        if (tid < off) red[tid] += red[tid + off];
        __syncthreads();
    }
    if (tid == 0) score[b] = red[0];
}

// ---------------------------------------------------------------------------
// Phase 2: forward scan in rescaled probability space using fp32 WMMA.
// Block = 128 threads (4 waves). Block bt owns batch rows [16*bt, 16*bt+16).
// Wave nt owns tag columns [16*nt, 16*nt+16). Per step, per wave:
//   D(16x16) = P(16x64) * E(64x16) via 16 chained v_wmma_f32_16x16x4_f32.
// Emission staging is software-pipelined one step ahead so the HBM load
// latency hides behind the WMMA chain + barriers instead of blocking them.
// ---------------------------------------------------------------------------
__global__ void __launch_bounds__(128)
crf_forward_scan(const float* __restrict__ labE,
                 const float* __restrict__ patE,
                 const int* __restrict__ t2l,
                 const float* __restrict__ tp,
                 const unsigned char* __restrict__ startMask,
                 const float* __restrict__ E,
                 const float* __restrict__ score,
                 float* __restrict__ out) {
    const int bt  = blockIdx.x;        // batch tile (16 of them)
    const int tid = threadIdx.x;
    const int nt  = tid >> 5;          // wave id == tag column tile
    const int l   = tid & 31;          // lane in wave32
    const int hh  = l >> 4;            // half-wave
    const int lm  = l & 15;

    const int n = nt * 16 + lm;        // this lane's tag column (C/D "N")

    __shared__ float sP[NTAGS * 17];               // p[tag][m], stride 17 (conflict-free)
    __shared__ __align__(16) float sLab[16 * 36];  // label emissions [m][32], stride 36
    __shared__ __align__(16) float sPat[16 * NPAT];
    __shared__ float sMaxP[4 * 16];                // per-tile row max [nt][m]
    __shared__ float sC[16];                       // accumulated log offset c[m]

    // Per-lane constants (tag column n is fixed for this lane).
    const int   t2l_n = t2l[n];
    const float tpn0 = tp[n * 4 + 0], tpn1 = tp[n * 4 + 1];
    const float tpn2 = tp[n * 4 + 2], tpn3 = tp[n * 4 + 3];
    const bool  start_n = startMask[n] != 0;

    // E column tile for this wave, resident in VGPRs as WMMA B operands.
    // f32 B 4x16 layout: VGPR0 = K=4kk   (lanes 0-15) / K=4kk+2 (lanes 16-31)
    //                    VGPR1 = K=4kk+1 (lanes 0-15) / K=4kk+3 (lanes 16-31)
    v2f eb[16];
#pragma unroll
    for (int kk = 0; kk < 16; ++kk) {
        int k0 = 4 * kk + 2 * hh;
        eb[kk][0] = E[(k0)     * NTAGS + n];
        eb[kk][1] = E[(k0 + 1) * NTAGS + n];
    }

    if (tid < 16) sC[tid] = 0.f;

    // Per-thread staging assignment: 16 rows x 8 float4 of label emissions,
    // threads 0..15 also carry the 4-float pattern row as one float4.
    const int    sm_ = tid >> 3, sj = tid & 7;
    const float* labBase = labE + ((size_t)(bt * 16 + sm_)) * SEQ * NLAB + 4 * sj;
    const float* patBase = patE + ((size_t)(bt * 16 + (tid & 15))) * SEQ * NPAT;

    // Prologue: prefetch step 0.
    float4 labReg = *(const float4*)(labBase);
    float4 patReg;
    if (tid < 16) patReg = *(const float4*)(patBase);

    __syncthreads();

    for (int s = 0; s < SEQ; ++s) {
        // ---- D = P * E (skip at s==0) -- issued before any loadcnt wait ----
        v8f d = {};
        if (s > 0) {
#pragma unroll
            for (int kk = 0; kk < 16; ++kk) {
                int k0 = 4 * kk + 2 * hh;
                v2f a;  // f32 A 16x4 layout: VGPR0 K=4kk/4kk+2, VGPR1 K=+1/+3
                a[0] = sP[(k0)     * 17 + lm];
                a[1] = sP[(k0 + 1) * 17 + lm];
                d = __builtin_amdgcn_wmma_f32_16x16x4_f32(
                        false, a, false, eb[kk], (short)0, d, false, false);
            }
        }

        // ---- commit this step's prefetched emissions to LDS ----
        *(float4*)&sLab[sm_ * 36 + 4 * sj] = labReg;
        if (tid < 16) *(float4*)&sPat[tid * 4] = patReg;

        // ---- prefetch next step (clamped; result unused at s==SEQ-1) ----
        {
            int sp = (s + 1 < SEQ) ? (s + 1) : s;
            labReg = *(const float4*)(labBase + (size_t)sp * NLAB);
            if (tid < 16) patReg = *(const float4*)(patBase + (size_t)sp * NPAT);
        }
        __syncthreads();  // staging visible; all waves done reading sP

        // ---- alpha' = log(D) + x  (D VGPR r holds rows m=r / m=r+8) ----
        float xv[8];
#pragma unroll
        for (int r = 0; r < 8; ++r) {
            int m = r + 8 * hh;
            xv[r] = sLab[m * 36 + t2l_n]
                  + sPat[m * 4 + 0] * tpn0 + sPat[m * 4 + 1] * tpn1
                  + sPat[m * 4 + 2] * tpn2 + sPat[m * 4 + 3] * tpn3;
        }
        float val[8];
        if (s == 0) {
#pragma unroll
            for (int r = 0; r < 8; ++r) val[r] = start_n ? xv[r] : NEG_INF;
        } else {
            // Raw v_log_f32: after renormalization each D-row max is >= ~e^-4,
            // so denormal inputs only occur for ~e^-100-weight entries; flushing
            // them to -inf (p -> 0) is harmless.
#pragma unroll
            for (int r = 0; r < 8; ++r) val[r] = hw_log2(d[r]) * LN2F + xv[r];
        }

        // ---- per-tile row maxima: butterfly within each 16-lane half ----
#pragma unroll
        for (int r = 0; r < 8; ++r) {
            float rm = val[r];
            rm = fmaxf(rm, __shfl_xor(rm, 1, 32));
            rm = fmaxf(rm, __shfl_xor(rm, 2, 32));
            rm = fmaxf(rm, __shfl_xor(rm, 4, 32));
            rm = fmaxf(rm, __shfl_xor(rm, 8, 32));
            if (lm == 0) sMaxP[nt * 16 + r + 8 * hh] = rm;
        }
        __syncthreads();  // sMaxP visible

        // ---- combine maxima across the 4 tag tiles, renormalize, store p ----
#pragma unroll
        for (int r = 0; r < 8; ++r) {
            int m = r + 8 * hh;
            float rm = fmaxf(fmaxf(sMaxP[m], sMaxP[16 + m]),
                             fmaxf(sMaxP[32 + m], sMaxP[48 + m]));
            sP[n * 17 + m] = hw_exp2((val[r] - rm) * INVLN2);
        }
        if (tid < 16) {
            float rm = fmaxf(fmaxf(sMaxP[tid], sMaxP[16 + tid]),
                             fmaxf(sMaxP[32 + tid], sMaxP[48 + tid]));
            sC[tid] += rm;
        }
        __syncthreads();  // sP / sC visible for next step
    }

    // ---- logZ(b) = c + log(sum_n p) ; out = score - logZ ----
    if (tid < 16) {
        float ssum = 0.f;
#pragma unroll
        for (int nn = 0; nn < NTAGS; ++nn) ssum += sP[nn * 17 + tid];
        float logZ = sC[tid] + hw_log2(ssum) * LN2F;
        int b = bt * 16 + tid;
        out[b] = score[b] - logZ;
    }
}

// ---------------------------------------------------------------------------
extern "C" void kernel_launch(void* const* d_in, const int* in_sizes, int n_in,
                              void* d_out, int out_size, void* d_ws, size_t ws_size,
                              hipStream_t stream) {
    const float*         labE = (const float*)d_in[0];          // [B,S,32]
    const float*         patE = (const float*)d_in[1];          // [B,S,4]
    const int*           y    = (const int*)d_in[2];            // [B,S]
    const float*         lt   = (const float*)d_in[3];          // [32,32]
    const int*           t2l  = (const int*)d_in[4];            // [64]
    const float*         tp   = (const float*)d_in[5];          // [64,4]
    const float*         tc   = (const float*)d_in[6];          // [64,64]
    const unsigned char* sm   = (const unsigned char*)d_in[7];  // [64] bool

    float* ws    = (float*)d_ws;
    float* E     = ws;          // 4096 floats
    float* T     = ws + 4096;   // 4096 floats
    float* score = ws + 8192;   // 256 floats

    crf_build_trans<<<16, 256, 0, stream>>>(lt, t2l, tc, E, T);
    crf_score<<<BATCH, 256, 0, stream>>>(labE, patE, y, t2l, tp, T, score);
    crf_forward_scan<<<16, 128, 0, stream>>>(labE, patE, t2l, tp, sm, E, score,
                                             (float*)d_out);
}